// TernaryTree1D_55903294325338
// MI455X (gfx1250) — compile-verified
//
#include <hip/hip_runtime.h>
#include <stdint.h>

// ---------------- problem constants ----------------
#define BATCH      16384
#define N_LEAVES   6561          // 3^8
#define N_INTERNAL 3280          // (3^8-1)/2
#define EPSF       1e-6f
#define CAPF       1000000.0f

// d_out layout (float32):
//   [0, 32768)        root complex64 as interleaved (re,im) per batch elem
//   [32768, 45890)    leaf_probs   (6561 x 2)
//   [45890, 75410)    gate_probs   (3280 x 3 x 3)
#define LEAF_OUT (2 * BATCH)
#define GATE_OUT (2 * BATCH + 2 * N_LEAVES)

// d_ws layout (float4 units):
//   nodeTab   : [0, 9840)        per (node,child): {p0*gcRe, p0*gcIm, p1, p2}
//   bottomTab : [9840, 16401)    leaf level folded into level-0 child params:
//                                {cRe + pc*aRe, cIm + pc*aIm, px + pc*w1, 0}
//   subRes    : byte 262416 ...  float2[27*16384], depth-5 subtree results
#define BOTTOM_F4    (3 * N_INTERNAL)
#define SUBRES_BYTE  ((3 * N_INTERNAL + N_LEAVES) * 16)

// node offsets per tree level (bottom-up): {0,2187,2916,3159,3240,3267,3276,3279}

constexpr int pow3(int n) { return n == 0 ? 1 : 3 * pow3(n - 1); }

// ---------------- math helpers ----------------
__device__ __forceinline__ float2 blend3(float4 p, float xb, float2 ch) {
    // p = {cRe, cIm, px, pc};  blend: c + px*x + pc*child   (x is real)
    return make_float2(fmaf(p.w, ch.x, fmaf(p.z, xb, p.x)),
                       fmaf(p.w, ch.y, p.y));
}

__device__ __forceinline__ float2 t_clamped(float2 A, float2 B, float2 C) {
    float nr = A.x * B.x - A.y * B.y;            // num = A*B
    float ni = A.x * B.y + A.y * B.x;
    float rr = fmaf(ni, C.y, nr * C.x);          //       * conj(C)
    float ri = fmaf(-nr, C.y, ni * C.x);
    float den = fmaf(C.x, C.x, fmaf(C.y, C.y, EPSF));
    // out = num/den clamped to |out|<=CAP  ==  num * CAP/max(|num|, CAP*den)
    // v_sqrt_f32 / v_rcp_f32 (~1 ulp each): ~2e-7 rel error per node, operands
    // bounded (den<=~1e12, CAP*den<=1e18, |num|^2<=~1e36) -> no overflow/denorms.
    float mag = __builtin_amdgcn_sqrtf(fmaf(rr, rr, ri * ri));
    float f   = CAPF * __builtin_amdgcn_rcpf(fmaxf(mag, CAPF * den));
    return make_float2(rr * f, ri * f);
}

// ---------------- kernel 1 (fused prep, cold, precise math) ----------------
// Gate entries [0, 6561) are exactly the level-0 (node,child) pairs, and leaf i
// pairs 1:1 with gate entry i. One kernel therefore produces gate_probs output,
// nodeTab, leaf_probs output, and the folded bottomTab without any intermediate
// global round-trip:  level-0 blend input = c + px*x + pc*(a + w1*x)
//                                         = (c + pc*a) + (px + pc*w1)*x
__global__ void __launch_bounds__(256) k_prep(const float* __restrict__ leaf_logits,
                                              const float* __restrict__ leaf_c,
                                              const float* __restrict__ gate_logits,
                                              const float* __restrict__ gate_c,
                                              float* __restrict__ out,
                                              float4* __restrict__ nodeTab,
                                              float4* __restrict__ bottomTab) {
    int i = blockIdx.x * 256 + threadIdx.x;        // i = node*3 + child
    if (i >= N_INTERNAL * 3) return;

    // ---- gate softmax (output 3) + node table ----
    float g0 = gate_logits[3 * i], g1 = gate_logits[3 * i + 1], g2 = gate_logits[3 * i + 2];
    float gm = fmaxf(g0, fmaxf(g1, g2));
    float ge0 = expf(g0 - gm), ge1 = expf(g1 - gm), ge2 = expf(g2 - gm);
    float ginv = 1.0f / (ge0 + ge1 + ge2);
    float p0 = ge0 * ginv, p1 = ge1 * ginv, p2 = ge2 * ginv;
    out[GATE_OUT + 3 * i]     = p0;
    out[GATE_OUT + 3 * i + 1] = p1;
    out[GATE_OUT + 3 * i + 2] = p2;
    float gcr = gate_c[2 * i], gci = gate_c[2 * i + 1];
    float4 nd = make_float4(p0 * gcr, p0 * gci, p1, p2);
    nodeTab[i] = nd;

    if (i < N_LEAVES) {
        // ---- leaf softmax (output 2) ----
        float l0 = leaf_logits[2 * i], l1 = leaf_logits[2 * i + 1];
        float lm = fmaxf(l0, l1);
        float le0 = expf(l0 - lm), le1 = expf(l1 - lm);
        float linv = 1.0f / (le0 + le1);
        float q0 = le0 * linv, q1 = le1 * linv;
        out[LEAF_OUT + 2 * i]     = q0;
        out[LEAF_OUT + 2 * i + 1] = q1;
        // ---- fold leaf into level-0 child params, all in registers ----
        float lcr = leaf_c[2 * i], lci = leaf_c[2 * i + 1];
        float aRe = q0 * lcr, aIm = q0 * lci;      // w0*c_leaf
        bottomTab[i] = make_float4(fmaf(nd.w, aRe, nd.x),
                                   fmaf(nd.w, aIm, nd.y),
                                   fmaf(nd.w, q1, nd.z), 0.0f);
    }
}

// ---------------- compile-time post-order walk over a depth-5 subtree ----------------
// prm (f4): [0,243) folded level-0 child params (81 nodes x 3);
//           [243,324) lvl1 (27x3); [324,351) lvl2 (9x3); [351,360) lvl3 (3x3); [360,363) lvl4 (1x3)
constexpr int SEGL(int l) { return l == 1 ? 243 : l == 2 ? 324 : l == 3 ? 351 : 360; }

template <int LVL, int M0>
__device__ __forceinline__ float2 walkSub(const float4* prm, float xb) {
    if constexpr (LVL == 0) {
        float4 q0 = prm[3 * M0 + 0];               // immediate-offset LDS broadcast
        float4 q1 = prm[3 * M0 + 1];
        float4 q2 = prm[3 * M0 + 2];
        float2 A = make_float2(fmaf(q0.z, xb, q0.x), q0.y);
        float2 B = make_float2(fmaf(q1.z, xb, q1.x), q1.y);
        float2 C = make_float2(fmaf(q2.z, xb, q2.x), q2.y);
        return t_clamped(A, B, C);
    } else {
        constexpr int STEP = pow3(LVL - 1);
        float2 a = walkSub<LVL - 1, M0>(prm, xb);
        float2 b = walkSub<LVL - 1, M0 + STEP>(prm, xb);
        float2 c = walkSub<LVL - 1, M0 + 2 * STEP>(prm, xb);
        constexpr int base = SEGL(LVL) + (M0 / pow3(LVL)) * 3;
        return t_clamped(blend3(prm[base + 0], xb, a),
                         blend3(prm[base + 1], xb, b),
                         blend3(prm[base + 2], xb, c));
    }
}

// ---------------- kernel 2: depth-5 subtree (hot kernel) ----------------
// grid = 27 * (BATCH/256); subtree s = blockIdx >> 6 (uniform), b-tile = blockIdx & 63
__global__ void __launch_bounds__(256) k_tree_sub(const float*  __restrict__ x,
                                                  const float4* __restrict__ bottomTab,
                                                  const float4* __restrict__ nodeTab,
                                                  float2*       __restrict__ subRes) {
    __shared__ __align__(16) float4 prm[363];      // 5.8 KB / block

    const int tid = threadIdx.x;
    const int s   = blockIdx.x >> 6;               // 0..26
    const int b   = ((blockIdx.x & 63) << 8) | tid;

    // ---- CDNA5 async global->LDS staging of this subtree's parameter slice ----
    for (int e = tid; e < 363; e += 256) {
        uint32_t dst = (uint32_t)(uintptr_t)(&prm[e]);     // low 32 bits = LDS offset
        if (e < 243) {
            uint32_t src = (uint32_t)((s * 243 + e) * 16);
            asm volatile("global_load_async_to_lds_b128 %0, %1, %2 offset:0"
                         :: "v"(dst), "v"(src), "s"((uint64_t)(uintptr_t)bottomTab)
                         : "memory");
        } else {
            int e2 = e - 243;
            int f4;
            if      (e2 <  81) f4 = (2187 + s * 27) * 3 +  e2;
            else if (e2 < 108) f4 = (2916 + s *  9) * 3 + (e2 -  81);
            else if (e2 < 117) f4 = (3159 + s *  3) * 3 + (e2 - 108);
            else               f4 = (3240 + s     ) * 3 + (e2 - 117);
            uint32_t src = (uint32_t)(f4 * 16);
            asm volatile("global_load_async_to_lds_b128 %0, %1, %2 offset:0"
                         :: "v"(dst), "v"(src), "s"((uint64_t)(uintptr_t)nodeTab)
                         : "memory");
        }
    }
    asm volatile("s_wait_asynccnt 0x0" ::: "memory");
    __syncthreads();

    const float xb = x[b];
    subRes[s * BATCH + b] = walkSub<4, 0>(prm, xb);        // tier-5 subtree value
}

// ---------------- compile-time walk for the top 3 levels (27 -> 1) ----------------
template <int LVL, int S0>
__device__ __forceinline__ float2 walkTop(const float4* __restrict__ nodeTab,
                                          const float2* __restrict__ subRes,
                                          int b, float xb) {
    if constexpr (LVL == 0) {
        return subRes[S0 * BATCH + b];
    } else {
        constexpr int STEP = pow3(LVL - 1);
        float2 a = walkTop<LVL - 1, S0>(nodeTab, subRes, b, xb);
        float2 bb = walkTop<LVL - 1, S0 + STEP>(nodeTab, subRes, b, xb);
        float2 c = walkTop<LVL - 1, S0 + 2 * STEP>(nodeTab, subRes, b, xb);
        constexpr int OFF  = (LVL == 1) ? 3267 : (LVL == 2) ? 3276 : 3279;
        constexpr int node = OFF + S0 / pow3(LVL);         // compile-time constant
        return t_clamped(blend3(nodeTab[node * 3 + 0], xb, a),
                         blend3(nodeTab[node * 3 + 1], xb, bb),
                         blend3(nodeTab[node * 3 + 2], xb, c));
    }
}

__global__ void __launch_bounds__(256) k_tree_top(const float*  __restrict__ x,
                                                  const float4* __restrict__ nodeTab,
                                                  const float2* __restrict__ subRes,
                                                  float2*       __restrict__ outRoot) {
    const int b    = blockIdx.x * 256 + threadIdx.x;
    const float xb = x[b];
    outRoot[b] = walkTop<3, 0>(nodeTab, subRes, b, xb);    // root value
}

// ---------------- launcher ----------------
extern "C" void kernel_launch(void* const* d_in, const int* in_sizes, int n_in,
                              void* d_out, int out_size, void* d_ws, size_t ws_size,
                              hipStream_t stream) {
    const float* x           = (const float*)d_in[0];
    const float* leaf_logits = (const float*)d_in[1];
    const float* leaf_c      = (const float*)d_in[2];
    const float* gate_logits = (const float*)d_in[3];
    const float* gate_c      = (const float*)d_in[4];
    float* out = (float*)d_out;

    float4* nodeTab   = (float4*)d_ws;
    float4* bottomTab = (float4*)d_ws + BOTTOM_F4;
    float2* subRes    = (float2*)((char*)d_ws + SUBRES_BYTE);   // needs ~3.8 MB of ws

    k_prep<<<(N_INTERNAL * 3 + 255) / 256, 256, 0, stream>>>(
        leaf_logits, leaf_c, gate_logits, gate_c, out, nodeTab, bottomTab);
    k_tree_sub<<<27 * (BATCH / 256), 256, 0, stream>>>(x, bottomTab, nodeTab, subRes);
    k_tree_top<<<BATCH / 256,        256, 0, stream>>>(x, nodeTab, subRes, (float2*)out);
}